// MSNmodel_30253749633308
// MI455X (gfx1250) — compile-verified
//
#include <hip/hip_runtime.h>

#define NBATCH 32
#define NPTS   32768
#define NPOINT 4096
#define NTHR   1024
#define KPT    32            // NPTS / NTHR points per thread

typedef __attribute__((ext_vector_type(2))) float f32x2;

// LDS layout (bytes): dist[N] | pz[N] | red[32 u64] | bc{idx,cx,cy,cz}
#define DIST_OFF   0
#define PZ_OFF     (NPTS * 4)              // 131072
#define RED_OFF    (PZ_OFF + NPTS * 4)     // 262144
#define BC_OFF     (RED_OFF + 32 * 8)      // 262400
#define SMEM_BYTES (BC_OFF + 32)           // 262432  (<= 320KB WGP LDS)

__global__ __launch_bounds__(NTHR, 1)
void fps_kernel(const float* __restrict__ xyz,
                int* __restrict__ oidx,
                float* __restrict__ oxyz)
{
    extern __shared__ unsigned char smem[];
    float*              distL = (float*)(smem + DIST_OFF);
    float*              pzL   = (float*)(smem + PZ_OFF);
    unsigned long long* redL  = (unsigned long long*)(smem + RED_OFF);
    int*                bcI   = (int*)(smem + BC_OFF);        // [0] = centroid idx
    float*              bcF   = (float*)(smem + BC_OFF + 4);  // cx, cy, cz

    const int b   = blockIdx.x;
    const int tid = threadIdx.x;
    const float* xb = xyz + (size_t)b * NPTS * 3;

    // x,y coordinates, two points per f32x2
    f32x2 px2[KPT / 2], py2[KPT / 2];

    // ---- Stage xyz into LDS with CDNA5 async global->LDS DMA ----
    // 4 chunks of 8192 points (96KB each), staged into the (not yet live)
    // dist region, then scattered: x,y -> VGPRs, z -> pz LDS plane.
    const unsigned stage_base = (unsigned)(unsigned long long)(void*)distL;
    #pragma unroll
    for (int c = 0; c < 4; ++c) {
        const unsigned char* gsrc = (const unsigned char*)(xb + (size_t)c * 8192 * 3);
        #pragma unroll
        for (int q = 0; q < 6; ++q) {
            unsigned loff = stage_base + (unsigned)(tid * 16 + q * 16384);
            const unsigned char* g = gsrc + tid * 16 + q * 16384;
            asm volatile("global_load_async_to_lds_b128 %0, %1, off"
                         :: "v"(loff), "v"(g) : "memory");
        }
        asm volatile("s_wait_asynccnt 0" ::: "memory");
        __syncthreads();
        const float* stage = (const float*)distL;
        #pragma unroll
        for (int kk = 0; kk < 8; ++kk) {
            const int k = c * 8 + kk;
            const int p = tid + kk * NTHR;     // index within chunk
            const int j = tid + k * NTHR;      // global point index
            px2[k >> 1][k & 1] = stage[3 * p + 0];
            py2[k >> 1][k & 1] = stage[3 * p + 1];
            pzL[j]             = stage[3 * p + 2];
        }
        __syncthreads();
    }

    // ---- init distance = 1e10, seed centroid = index 1 ----
    #pragma unroll
    for (int k = 0; k < KPT; ++k)
        distL[tid + k * NTHR] = 1.0e10f;
    if (tid == 1) {                 // thread 1 owns point j=1 (k=0)
        bcI[0] = 1;
        bcF[0] = px2[0][0];
        bcF[1] = py2[0][0];
        bcF[2] = pzL[1];
    }
    __syncthreads();

    // ---- 4096 sequential FPS steps, fully WGP-resident ----
    for (int i = 0; i < NPOINT; ++i) {
        const int   cidx = bcI[0];
        const float cx = bcF[0], cy = bcF[1], cz = bcF[2];

        if (tid == 0) {  // emit idx + gathered coords for this step
            oidx[b * NPOINT + i] = cidx;
            float* o = oxyz + ((size_t)b * NPOINT + i) * 3;
            o[0] = cx; o[1] = cy; o[2] = cz;
        }

        const f32x2 cx2 = {cx, cx};
        const f32x2 cy2 = {cy, cy};
        const f32x2 cz2 = {cz, cz};

        // local min-update + argmax tracking; chunks of 8 so DS ops batch.
        // dist update is one ds_min_num_rtn_f32 per point.
        float bestv = -1.0f;
        int   bestk = 0;
        #pragma unroll
        for (int c = 0; c < KPT / 8; ++c) {
            float pzv[8];
            #pragma unroll
            for (int u = 0; u < 8; ++u)
                pzv[u] = pzL[tid + (c * 8 + u) * NTHR];

            float dsq[8];
            #pragma unroll
            for (int pp = 0; pp < 4; ++pp) {
                const int P = c * 4 + pp;
                f32x2 pz2;
                pz2[0] = pzv[2 * pp];
                pz2[1] = pzv[2 * pp + 1];
                const f32x2 dx = px2[P] - cx2;
                const f32x2 dy = py2[P] - cy2;
                const f32x2 dz = pz2 - cz2;
                const f32x2 d2 = dx * dx + dy * dy + dz * dz;
                dsq[2 * pp]     = d2[0];
                dsq[2 * pp + 1] = d2[1];
            }

            float oldd[8];
            #pragma unroll
            for (int u = 0; u < 8; ++u) {
                const int j = tid + (c * 8 + u) * NTHR;
                oldd[u] = __hip_atomic_fetch_min(&distL[j], dsq[u],
                                                 __ATOMIC_RELAXED,
                                                 __HIP_MEMORY_SCOPE_WORKGROUP);
            }

            #pragma unroll
            for (int u = 0; u < 8; ++u) {
                const int k = c * 8 + u;
                const float nd = fminf(oldd[u], dsq[u]);
                const bool gt = nd > bestv;       // strict: earliest k on tie
                bestk = gt ? k : bestk;           // k is an inline constant
                bestv = gt ? nd : bestv;
            }
        }

        // packed key: high = dist bits (>=0 so order-monotonic),
        // low = N-1-j (so ties pick the smallest index)
        const int bestj = tid + (bestk << 10);
        unsigned long long best =
            ((unsigned long long)__float_as_uint(bestv) << 32) |
            (unsigned)(NPTS - 1 - bestj);

        // wave32 shuffle max-reduce
        #pragma unroll
        for (int off = 16; off >= 1; off >>= 1) {
            const unsigned long long o = __shfl_xor(best, off, 32);
            best = (o > best) ? o : best;
        }
        if ((tid & 31) == 0) redL[tid >> 5] = best;
        __syncthreads();

        // cross-wave reduce in wave 0
        if (tid < 32) {
            unsigned long long v = redL[tid];
            #pragma unroll
            for (int off = 16; off >= 1; off >>= 1) {
                const unsigned long long o = __shfl_xor(v, off, 32);
                v = (o > v) ? o : v;
            }
            if (tid == 0)
                bcI[0] = NPTS - 1 - (int)(unsigned)(v & 0xffffffffull);
        }
        __syncthreads();

        // owning thread publishes the new centroid's coordinates
        const int widx = bcI[0];
        if (tid == (widx & (NTHR - 1))) {
            const int ks = widx >> 10;        // log2(NTHR)
            const int pr = ks >> 1;
            f32x2 sx = px2[0], sy = py2[0];
            #pragma unroll
            for (int p = 1; p < KPT / 2; ++p) {
                sx = (p == pr) ? px2[p] : sx;
                sy = (p == pr) ? py2[p] : sy;
            }
            const int e = ks & 1;
            bcF[0] = e ? sx[1] : sx[0];
            bcF[1] = e ? sy[1] : sy[0];
            bcF[2] = pzL[widx];
        }
        __syncthreads();
    }
}

extern "C" void kernel_launch(void* const* d_in, const int* in_sizes, int n_in,
                              void* d_out, int out_size, void* d_ws, size_t ws_size,
                              hipStream_t stream) {
    (void)in_sizes; (void)n_in; (void)d_ws; (void)ws_size; (void)out_size;
    const float* xyz = (const float*)d_in[0];     // [32, 32768, 3] f32
    // d_in[1] is npoint (==4096, fixed by the reference setup)
    int*   oidx = (int*)d_out;                          // 32*4096 int32 (bit view)
    float* oxyz = (float*)d_out + NBATCH * NPOINT;      // 32*4096*3 f32

    hipFuncSetAttribute(reinterpret_cast<const void*>(fps_kernel),
                        hipFuncAttributeMaxDynamicSharedMemorySize, SMEM_BYTES);
    fps_kernel<<<dim3(NBATCH), dim3(NTHR), SMEM_BYTES, stream>>>(xyz, oidx, oxyz);
}